// Pseudo_random_Interleaver_67465346286283
// MI455X (gfx1250) — compile-verified
//
#include <hip/hip_runtime.h>

// Pseudo-random interleaver: out[b, i] = x[b, perm[b, i]]
//   x:    [B, L] float32  (B = 1024, L = 16384)
//   perm: [B, L] int32    (row-wise permutation, values in [0, L))
//
// Strategy (MI455X / gfx1250):
//  - Pure memory-bound gather: ~192 MiB traffic -> ~8.6 us @ 23.3 TB/s.
//  - One workgroup per row. Stage the 64 KiB row into LDS with coalesced
//    128-bit ASYNC global->LDS copies (CDNA5 async path, ASYNCcnt), then
//    gather from LDS so the random access never touches global memory.
//  - perm read and out write are 128-bit coalesced.

#define LROW 16384
#define BLK  256
#define V4_PER_ROW   (LROW / 4)          // 4096 float4 per row
#define V4_PER_THRD  (V4_PER_ROW / BLK)  // 16 float4 per thread

typedef int v4i __attribute__((vector_size(16)));  // pointer type the builtin expects

__global__ __launch_bounds__(BLK) void interleaver_gather_kernel(
    const float* __restrict__ x,
    const int*   __restrict__ perm,
    float*       __restrict__ out)
{
    __shared__ float row[LROW];  // 64 KiB: full x-row staged in LDS

    const int    b   = blockIdx.x;
    const int    tid = threadIdx.x;
    const size_t base = (size_t)b * (size_t)LROW;

    const float* xrow = x    + base;
    const int*   prow = perm + base;
    float*       orow = out  + base;

    // ---- Stage 1: coalesced 128-bit copy of the row into LDS ----
#if __has_builtin(__builtin_amdgcn_global_load_async_to_lds_b128)
    #pragma unroll
    for (int i = 0; i < V4_PER_THRD; ++i) {
        const int idx4 = tid + i * BLK;                   // float4 index
        __builtin_amdgcn_global_load_async_to_lds_b128(
            (v4i*)(xrow + 4 * idx4),
            (v4i*)(row  + 4 * idx4),
            /*offset=*/0, /*cpol=*/0);
    }
    // Wait for this wave's async transfers; barrier covers the other waves.
  #if __has_builtin(__builtin_amdgcn_s_wait_asynccnt)
    __builtin_amdgcn_s_wait_asynccnt(0);
  #else
    asm volatile("s_wait_asynccnt 0" ::: "memory");
  #endif
#else
    // Fallback: classic VGPR-bounced copy (global_load_b128 + ds_store_b128)
    #pragma unroll
    for (int i = 0; i < V4_PER_THRD; ++i) {
        const int idx4 = tid + i * BLK;
        float4 v = ((const float4*)xrow)[idx4];
        ((float4*)row)[idx4] = v;
    }
#endif
    __syncthreads();

    // ---- Stage 2: coalesced perm read, LDS gather, coalesced store ----
    #pragma unroll
    for (int i = 0; i < V4_PER_THRD; ++i) {
        const int idx4 = tid + i * BLK;
        const int4 p = ((const int4*)prow)[idx4];   // 128-bit coalesced
        float4 o;
        o.x = row[p.x];                             // random ds_load_b32
        o.y = row[p.y];
        o.z = row[p.z];
        o.w = row[p.w];
        ((float4*)orow)[idx4] = o;                  // 128-bit coalesced
    }
}

extern "C" void kernel_launch(void* const* d_in, const int* in_sizes, int n_in,
                              void* d_out, int out_size, void* d_ws, size_t ws_size,
                              hipStream_t stream)
{
    const float* x    = (const float*)d_in[0];   // [B, L, 1] float32
    const int*   perm = (const int*)  d_in[1];   // [B, L]    int32
    float*       out  = (float*)d_out;           // [B, L, 1] float32

    const int B = in_sizes[0] / LROW;            // 1024
    interleaver_gather_kernel<<<dim3(B), dim3(BLK), 0, stream>>>(x, perm, out);
}